// Hilbert3DGetIdx_32899449487894
// MI455X (gfx1250) — compile-verified
//
#include <hip/hip_runtime.h>
#include <hip/hip_bf16.h>
#include <stdint.h>

// Problem constants (reference shapes are fixed: z is [32, 131072, 3] f32).
#define B_      32
#define N_      131072
#define NB_     512           // radix tiles per batch (tile = 256 keys)
#define RADIX   256
#define PASSES  8             // 8 x 8-bit digits covers the 63-bit Hilbert key

#if __has_builtin(__builtin_amdgcn_global_load_async_to_lds_b64)
#define HAVE_ASYNC_LDS 1
#else
#define HAVE_ASYNC_LDS 0
#endif

// Builtin signature (from hipcc diagnostic): pointee is int __vector_size__(8)
typedef int v2i_ __attribute__((vector_size(8)));
typedef __attribute__((address_space(1))) v2i_* g_v2i_ptr;
typedef __attribute__((address_space(3))) v2i_* l_v2i_ptr;

__device__ __forceinline__ void wait_async_zero() {
#if __has_builtin(__builtin_amdgcn_s_wait_asynccnt)
  __builtin_amdgcn_s_wait_asynccnt(0);
#else
  asm volatile("s_wait_asynccnt 0" ::: "memory");
#endif
}

// ---------------- reductions ----------------

__global__ void k_init(double* sums, unsigned long long* maxb) {
  int t = threadIdx.x;
  if (t < B_ * 3) sums[t] = 0.0;
  if (t < B_)     maxb[t] = 0ull;
}

__global__ void k_sum(const float* __restrict__ z, double* sums) {
  int b   = blockIdx.x >> 6;     // 64 blocks per batch
  int sub = blockIdx.x & 63;
  const float* zb = z + (size_t)b * N_ * 3;
  double sx = 0.0, sy = 0.0, sz = 0.0;
  int base = sub * (N_ / 64);    // 2048 points per block
  for (int k = threadIdx.x; k < N_ / 64; k += 256) {
    int i = base + k;
    sx += (double)zb[(size_t)i * 3 + 0];
    sy += (double)zb[(size_t)i * 3 + 1];
    sz += (double)zb[(size_t)i * 3 + 2];
  }
  for (int o = 16; o > 0; o >>= 1) {
    sx += __shfl_down(sx, o, 32);
    sy += __shfl_down(sy, o, 32);
    sz += __shfl_down(sz, o, 32);
  }
  if ((threadIdx.x & 31) == 0) {
    atomicAdd(&sums[b * 3 + 0], sx);   // global_atomic_add_f64
    atomicAdd(&sums[b * 3 + 1], sy);
    atomicAdd(&sums[b * 3 + 2], sz);
  }
}

__global__ void k_max(const float* __restrict__ z, const double* __restrict__ sums,
                      unsigned long long* maxb) {
  int b   = blockIdx.x >> 6;
  int sub = blockIdx.x & 63;
  const float* zb = z + (size_t)b * N_ * 3;
  const double invN = 1.0 / (double)N_;
  double cx = sums[b * 3 + 0] * invN;
  double cy = sums[b * 3 + 1] * invN;
  double cz = sums[b * 3 + 2] * invN;
  double m = 0.0;
  int base = sub * (N_ / 64);
  for (int k = threadIdx.x; k < N_ / 64; k += 256) {
    int i = base + k;
    double dx = (double)zb[(size_t)i * 3 + 0] - cx;
    double dy = (double)zb[(size_t)i * 3 + 1] - cy;
    double dz = (double)zb[(size_t)i * 3 + 2] - cz;
    double d2 = dx * dx + dy * dy + dz * dz;
    m = fmax(m, d2);
  }
  for (int o = 16; o > 0; o >>= 1) m = fmax(m, __shfl_down(m, o, 32));
  if ((threadIdx.x & 31) == 0) {
    // d2 >= 0 -> IEEE bit pattern is monotone as u64
    atomicMax(&maxb[b], (unsigned long long)__double_as_longlong(m));
  }
}

// ---------------- Hilbert encode ----------------

__device__ __forceinline__ unsigned long long spread21(unsigned long long v) {
  v &= 0x1FFFFFull;
  v = (v | (v << 32)) & 0x001F00000000FFFFull;
  v = (v | (v << 16)) & 0x001F0000FF0000FFull;
  v = (v | (v << 8))  & 0x100F00F00F00F00Full;
  v = (v | (v << 4))  & 0x10C30C30C30C30C3ull;
  v = (v | (v << 2))  & 0x1249249249249249ull;
  return v;
}

__global__ void k_encode(const float* __restrict__ z, const double* __restrict__ sums,
                         const unsigned long long* __restrict__ maxb,
                         unsigned long long* __restrict__ keys,
                         unsigned int* __restrict__ vals) {
  int b   = blockIdx.x >> 9;     // NB_=512 tiles per batch
  int blk = blockIdx.x & 511;
  int i   = blk * 256 + (int)threadIdx.x;
  const float* zp = z + ((size_t)b * N_ + i) * 3;
  __builtin_prefetch(zp, 0, 1);  // global_prefetch_b8
  const double invN = 1.0 / (double)N_;
  double cx = sums[b * 3 + 0] * invN;
  double cy = sums[b * 3 + 1] * invN;
  double cz = sums[b * 3 + 2] * invN;
  double r  = sqrt(__longlong_as_double((long long)maxb[b]));
  double zr = r + 1e-6;
  double inv = 1.0 / (2.0 * zr);

  unsigned long long X[3];
  {
    double n0 = ((double)zp[0] - cx + zr) * inv;
    double n1 = ((double)zp[1] - cy + zr) * inv;
    double n2 = ((double)zp[2] - cz + zr) * inv;
    X[0] = (unsigned long long)((long long)(1048576.0 * n0) + 32767ll);
    X[1] = (unsigned long long)((long long)(1048576.0 * n1) + 32767ll);
    X[2] = (unsigned long long)((long long)(1048576.0 * n2) + 32767ll);
  }
  // Skilling AxesToTranspose, num_bits = 21, M = 1<<20
  for (unsigned long long Q = (1ull << 20); Q > 1; Q >>= 1) {
    unsigned long long P = Q - 1;
#pragma unroll
    for (int d = 0; d < 3; ++d) {
      if (X[d] & Q) {
        X[0] ^= P;
      } else {
        unsigned long long t = (X[0] ^ X[d]) & P;
        X[0] ^= t;
        X[d] ^= t;
      }
    }
  }
  X[1] ^= X[0];
  X[2] ^= X[1];
  unsigned long long t = 0;
  for (unsigned long long Q = (1ull << 20); Q > 1; Q >>= 1)
    if (X[2] & Q) t ^= (Q - 1);
  X[0] ^= t; X[1] ^= t; X[2] ^= t;

  unsigned long long h = (spread21(X[0]) << 2) | (spread21(X[1]) << 1) | spread21(X[2]);
  keys[(size_t)b * N_ + i] = h;
  vals[(size_t)b * N_ + i] = (unsigned int)i;
}

// ---------------- radix sort (LSD, stable) ----------------

__global__ void k_hist(unsigned long long* __restrict__ src,
                       unsigned int* __restrict__ hist, int shift) {
  __shared__ unsigned long long tileK[256];
  __shared__ unsigned int lh[RADIX];
  int b   = blockIdx.x >> 9;
  int blk = blockIdx.x & 511;
  int tid = threadIdx.x;
  lh[tid] = 0u;
  size_t gi = (size_t)b * N_ + (size_t)blk * 256 + tid;
#if HAVE_ASYNC_LDS
  // CDNA5 async data path: global -> LDS without touching VGPRs (ASYNCcnt tracked)
  __builtin_amdgcn_global_load_async_to_lds_b64(
      (g_v2i_ptr)(src + gi),
      (l_v2i_ptr)(&tileK[tid]), 0, 0);
  wait_async_zero();
#else
  tileK[tid] = src[gi];
#endif
  __syncthreads();
  unsigned d = (unsigned)((tileK[tid] >> shift) & 255ull);
  atomicAdd(&lh[d], 1u);                 // ds_add_u32
  __syncthreads();
  hist[((size_t)b * NB_ + blk) * RADIX + tid] = lh[tid];
}

__global__ void k_scan(unsigned int* __restrict__ hist, unsigned int* __restrict__ dstart) {
  int b = blockIdx.x;      // one block per batch
  int d = threadIdx.x;     // one thread per digit
  __shared__ unsigned int tot[RADIX];
  unsigned int run = 0;
  size_t base = (size_t)b * NB_ * RADIX + d;
  for (int blk = 0; blk < NB_; ++blk) {
    size_t idx = base + (size_t)blk * RADIX;
    unsigned c = hist[idx];
    hist[idx] = run;        // in-place exclusive block prefix per digit
    run += c;
  }
  tot[d] = run;
  __syncthreads();
  if (d == 0) {
    unsigned run2 = 0;
    for (int k = 0; k < RADIX; ++k) { unsigned c = tot[k]; dstart[b * RADIX + k] = run2; run2 += c; }
  }
}

__global__ void k_scatter(const unsigned long long* __restrict__ srcK,
                          const unsigned int* __restrict__ srcV,
                          unsigned long long* __restrict__ dstK,
                          unsigned int* __restrict__ dstV,
                          const unsigned int* __restrict__ hist,
                          const unsigned int* __restrict__ dstart, int shift) {
  __shared__ unsigned int wh[8][RADIX];   // per-wave digit counts (wave32: 8 waves/block)
  int b   = blockIdx.x >> 9;
  int blk = blockIdx.x & 511;
  int tid = threadIdx.x;
#pragma unroll
  for (int w = 0; w < 8; ++w) wh[w][tid] = 0u;
  size_t gi = (size_t)b * N_ + (size_t)blk * 256 + tid;
  unsigned long long key = srcK[gi];
  unsigned int val = srcV[gi];
  unsigned dgt = (unsigned)((key >> shift) & 255ull);
  __syncthreads();

  int lane = tid & 31, wave = tid >> 5;
  // wave32 multi-split: find lanes sharing this digit via 8 ballots
  unsigned peers = 0xFFFFFFFFu;
#pragma unroll
  for (int bit = 0; bit < 8; ++bit) {
    unsigned v = (unsigned)__ballot((dgt >> bit) & 1u);
    peers &= ((dgt >> bit) & 1u) ? v : ~v;
  }
  unsigned rank = (unsigned)__popc(peers & ((1u << lane) - 1u));
  if (rank == 0) wh[wave][dgt] = (unsigned)__popc(peers);
  __syncthreads();
  unsigned before = 0;
  for (int w = 0; w < wave; ++w) before += wh[w][dgt];  // stable: earlier waves first
  unsigned dest = dstart[b * RADIX + dgt]
                + hist[((size_t)b * NB_ + blk) * RADIX + dgt]
                + before + rank;
  dstK[(size_t)b * N_ + dest] = key;
  dstV[(size_t)b * N_ + dest] = val;
}

__global__ void k_final(const unsigned int* __restrict__ vals, float* __restrict__ out) {
  int b   = blockIdx.x >> 9;
  int blk = blockIdx.x & 511;
  int i   = blk * 256 + (int)threadIdx.x;
  unsigned v = vals[(size_t)b * N_ + i];
  out[(size_t)b * N_ + i]                       = (float)v;  // idx_pa
  out[(size_t)B_ * N_ + (size_t)b * N_ + v]     = (float)i;  // idx_re (inverse perm)
}

// ---------------- host launcher ----------------

extern "C" void kernel_launch(void* const* d_in, const int* in_sizes, int n_in,
                              void* d_out, int out_size, void* d_ws, size_t ws_size,
                              hipStream_t stream) {
  (void)in_sizes; (void)n_in; (void)out_size; (void)ws_size;
  const float* z = (const float*)d_in[0];
  float* out = (float*)d_out;

  char* ws = (char*)d_ws;
  double*             sums  = (double*)ws;                          // 32*3 f64
  unsigned long long* maxb  = (unsigned long long*)(ws + 1024);     // 32 u64
  size_t off = 2048;
  unsigned long long* keysA = (unsigned long long*)(ws + off); off += (size_t)B_ * N_ * 8;
  unsigned long long* keysB = (unsigned long long*)(ws + off); off += (size_t)B_ * N_ * 8;
  unsigned int*       valsA = (unsigned int*)(ws + off);       off += (size_t)B_ * N_ * 4;
  unsigned int*       valsB = (unsigned int*)(ws + off);       off += (size_t)B_ * N_ * 4;
  unsigned int*       hist  = (unsigned int*)(ws + off);       off += (size_t)B_ * NB_ * RADIX * 4;
  unsigned int*       dst0  = (unsigned int*)(ws + off);       off += (size_t)B_ * RADIX * 4;

  k_init<<<1, 128, 0, stream>>>(sums, maxb);
  k_sum<<<B_ * 64, 256, 0, stream>>>(z, sums);
  k_max<<<B_ * 64, 256, 0, stream>>>(z, sums, maxb);
  k_encode<<<B_ * NB_, 256, 0, stream>>>(z, sums, maxb, keysA, valsA);

  unsigned long long* sk = keysA; unsigned long long* dk = keysB;
  unsigned int* sv = valsA;       unsigned int* dv = valsB;
  for (int p = 0; p < PASSES; ++p) {
    int shift = p * 8;
    k_hist<<<B_ * NB_, 256, 0, stream>>>(sk, hist, shift);
    k_scan<<<B_, 256, 0, stream>>>(hist, dst0);
    k_scatter<<<B_ * NB_, 256, 0, stream>>>(sk, sv, dk, dv, hist, dst0, shift);
    unsigned long long* tk = sk; sk = dk; dk = tk;
    unsigned int* tv = sv; sv = dv; dv = tv;
  }
  // PASSES is even -> final sorted payload back in valsA (== sv after loop)
  k_final<<<B_ * NB_, 256, 0, stream>>>(sv, out);
}